// SLDASNet_33921651704421
// MI455X (gfx1250) — compile-verified
//
#include <hip/hip_runtime.h>
#include <stdint.h>

// ---------------------------------------------------------------------------
// 1-D KNN (k=3) feature builder for MI455X / gfx1250.
//
// Strategy: x_measured (16 KB) is DMA'd into LDS with the Tensor Data Mover
// (CDNA5 TENSOR_LOAD_TO_LDS, tracked by TENSORcnt), bitonic-sorted in LDS,
// and then each query does a branchless lower_bound + 3-step sorted merge.
// Output per query: [x, d0, d1, d2] with d0<=d1<=d2 (matches sort+slice).
// ---------------------------------------------------------------------------

#define SORT_SZ 4096   // padded pow-2 LDS table (16 KB of the 320 KB/WGP LDS)
#define BLOCK_T 256    // 8 wave32 waves per workgroup
#define XPT     2      // queries per thread (512 queries per block)

typedef unsigned int u32x4 __attribute__((ext_vector_type(4)));
typedef int          i32x4 __attribute__((ext_vector_type(4)));
typedef int          i32x8 __attribute__((ext_vector_type(8)));

#if __has_builtin(__builtin_amdgcn_tensor_load_to_lds) && __has_builtin(__builtin_amdgcn_s_wait_tensorcnt)
#define HAVE_TDM 1
#else
#define HAVE_TDM 0
#endif

// DMA n fp32 elements global->LDS. TDM descriptor per CDNA5 ISA §8.3/8.4:
// group0: count=1, lds_addr, global_addr[56:0], type=2 ("image").
// group1: data_size=2 (4B), tensor_dim0=tile_dim0=n (1-D tile), stride0=n.
// TENSOR_* ops ignore EXEC; one wave issues, then waits TENSORcnt==0.
__device__ __forceinline__ void stage_f32_to_lds(float* lds_dst,
                                                 const float* gsrc,
                                                 int n, int tid, int nthreads) {
#if HAVE_TDM
  (void)nthreads;
  if (tid < 32) {  // wave 0 only (wave32)
    const uint64_t ga = (uint64_t)(uintptr_t)gsrc;
    const uint32_t la = (uint32_t)(uintptr_t)lds_dst;  // low 32b = LDS byte addr
    u32x4 g0;
    g0.x = 1u;                                                   // count = 1
    g0.y = la;                                                   // lds_addr
    g0.z = (uint32_t)ga;                                         // addr[31:0]
    g0.w = (uint32_t)((ga >> 32) & 0x01FFFFFFu) | 0x80000000u;   // addr[56:32]|type=2
    i32x8 g1;
    g1[0] = 0x00020000;                         // wg_mask=0, data_size=4B
    g1[1] = (int)((uint32_t)n << 16);           // tensor_dim0[15:0]
    g1[2] = (int)((uint32_t)n >> 16);           // tensor_dim0[31:16], tensor_dim1=0
    g1[3] = (int)((uint32_t)n << 16);           // tile_dim0 = n (1-D tile)
    g1[4] = 0;                                  // tile_dim1/2 unused
    g1[5] = n;                                  // tensor_dim0_stride (don't-care)
    g1[6] = 0;
    g1[7] = 0;
    i32x4 z4 = {0, 0, 0, 0};
#if defined(__clang_major__) && __clang_major__ >= 23
    i32x8 z8 = {0, 0, 0, 0, 0, 0, 0, 0};
    __builtin_amdgcn_tensor_load_to_lds(g0, g1, z4, z4, z8, 0);
#else
    __builtin_amdgcn_tensor_load_to_lds(g0, g1, z4, z4, 0);
#endif
    __builtin_amdgcn_s_wait_tensorcnt(0);
  }
#else
  for (int i = tid; i < n; i += nthreads) lds_dst[i] = gsrc[i];
#endif
}

// Main path (M <= SORT_SZ): per-block LDS sort + binary-search KNN.
__global__ __launch_bounds__(BLOCK_T) void knn_sorted_kernel(
    const float* __restrict__ x, const float* __restrict__ xm,
    float* __restrict__ out, int N, int M) {
  __shared__ float s[SORT_SZ];
  const int tid = threadIdx.x;
  const float FINF = __builtin_inff();

  // 1) Stage x_measured into LDS via TDM; pad tail with +inf for pow-2 sort.
  stage_f32_to_lds(s, xm, M, tid, BLOCK_T);
  for (int i = M + tid; i < SORT_SZ; i += BLOCK_T) s[i] = FINF;
  __syncthreads();

  // 2) In-LDS bitonic sort, ascending. 78 phases; pairs disjoint per phase.
  for (int k = 2; k <= SORT_SZ; k <<= 1) {
    for (int j = k >> 1; j > 0; j >>= 1) {
      for (int i = tid; i < SORT_SZ; i += BLOCK_T) {
        const int ixj = i ^ j;
        if (ixj > i) {
          const float a = s[i];
          const float b = s[ixj];
          const bool asc = ((i & k) == 0);
          if ((a > b) == asc) { s[i] = b; s[ixj] = a; }
        }
      }
      __syncthreads();
    }
  }

  // 3) Per query: branchless lower_bound (12 steps) + 3-step sorted merge.
  for (int v = 0; v < XPT; ++v) {
    const int gid = (blockIdx.x * XPT + v) * BLOCK_T + tid;
    if (gid >= N) continue;
    const float xv = x[gid];

    int lo = 0;  // lower_bound over pow-2 array
#pragma unroll
    for (int step = SORT_SZ >> 1; step > 0; step >>= 1) {
      if (s[lo + step - 1] < xv) lo += step;
    }

    int l = lo - 1, r = lo;
    float k0, k1, k2;
#pragma unroll
    for (int t = 0; t < 3; ++t) {
      const float dl = (l >= 0)      ? fabsf(xv - s[l]) : FINF;
      const float dr = (r < SORT_SZ) ? fabsf(s[r] - xv) : FINF;
      float d;
      if (dl <= dr) { d = dl; --l; } else { d = dr; ++r; }
      if (t == 0) k0 = d; else if (t == 1) k1 = d; else k2 = d;
    }
    // nan_to_num(nan=1.0) parity with the reference (no-op for finite inputs)
    k0 = (k0 == k0) ? k0 : 1.0f;
    k1 = (k1 == k1) ? k1 : 1.0f;
    k2 = (k2 == k2) ? k2 : 1.0f;

    reinterpret_cast<float4*>(out)[gid] = make_float4(xv, k0, k1, k2);
  }
}

// Fallback (M > SORT_SZ): chunked brute force, branchless 3-min network.
__global__ __launch_bounds__(BLOCK_T) void knn_brute_kernel(
    const float* __restrict__ x, const float* __restrict__ xm,
    float* __restrict__ out, int N, int M) {
  __shared__ float s[SORT_SZ];
  const int tid = threadIdx.x;
  const int gid = blockIdx.x * BLOCK_T + tid;
  const float xv = (gid < N) ? x[gid] : 0.0f;
  float b0 = 3.402823466e+38f, b1 = b0, b2 = b0;

  for (int base = 0; base < M; base += SORT_SZ) {
    const int len = min(SORT_SZ, M - base);
    stage_f32_to_lds(s, xm + base, len, tid, BLOCK_T);
    __syncthreads();
#pragma unroll 4
    for (int j = 0; j < len; ++j) {
      const float d = fabsf(xv - s[j]);       // LDS broadcast read (no conflicts)
      b2 = fminf(b2, fmaxf(d, b1));           // branchless sorted-insert network
      b1 = fminf(b1, fmaxf(d, b0));
      b0 = fminf(b0, d);
    }
    __syncthreads();
  }

  if (gid < N) {
    b0 = (b0 == b0) ? b0 : 1.0f;
    b1 = (b1 == b1) ? b1 : 1.0f;
    b2 = (b2 == b2) ? b2 : 1.0f;
    reinterpret_cast<float4*>(out)[gid] = make_float4(xv, b0, b1, b2);
  }
}

extern "C" void kernel_launch(void* const* d_in, const int* in_sizes, int n_in,
                              void* d_out, int out_size, void* d_ws, size_t ws_size,
                              hipStream_t stream) {
  (void)n_in; (void)out_size; (void)d_ws; (void)ws_size;
  const float* x  = (const float*)d_in[0];
  const float* xm = (const float*)d_in[1];
  float* out = (float*)d_out;
  const int N = in_sizes[0];
  const int M = in_sizes[1];

  if (M <= SORT_SZ) {
    const int per_block = BLOCK_T * XPT;
    const int grid = (N + per_block - 1) / per_block;   // 128 blocks for N=65536
    knn_sorted_kernel<<<grid, BLOCK_T, 0, stream>>>(x, xm, out, N, M);
  } else {
    const int grid = (N + BLOCK_T - 1) / BLOCK_T;
    knn_brute_kernel<<<grid, BLOCK_T, 0, stream>>>(x, xm, out, N, M);
  }
}